// Model_42253888258529
// MI455X (gfx1250) — compile-verified
//
#include <hip/hip_runtime.h>

// Lucas-Kanade optical flow, fused single kernel for MI455X (gfx1250, wave32).
// Memory-bound (64MB min traffic -> ~2.8us @ 23.3 TB/s; ~4us with halo amp).
// 15x15 box sums = banded-matrix WMMA f32 16x16x4 matmuls per 16x16 wave tile.
// Round 3: targeted pad-zeroing (780 cells instead of 12.8K), division-free
// incremental addressing in the fill/Sobel stages, hoisted column reflection.

typedef __attribute__((ext_vector_type(2))) float v2f;
typedef __attribute__((ext_vector_type(8))) float v8f;

#define H_IMG 2048
#define W_IMG 2048
#define PH 82              // plane rows (80 used + 2 zero-pad for WMMA M padding)
#define PW 52              // plane cols (50 used + pad; stride -> bank-conflict-free)
#define PP (PH * PW)
#define TS 36              // T row stride in dwords (16B-aligned rows, distinct banks)
#define TSZ (16 * TS)      // per-product T buffer (16 cols x 32 rows, transposed)
#define NT 256             // 8 waves of 32

__device__ __forceinline__ int refl(int g) {
  // jnp.pad reflect index map for [0, 2047]
  return g < 0 ? -g : (g > (H_IMG - 1) ? 2 * (H_IMG - 1) - g : g);
}

__device__ __forceinline__ float bandv(int k, int i) {
  // 15-tap box band matrix entry: 1 if k - i in [0, 14]
  return (unsigned)(k - i) <= 14u ? 1.0f : 0.0f;
}

__device__ __forceinline__ v8f wmma4(v2f a, v2f b, v8f c) {
  return __builtin_amdgcn_wmma_f32_16x16x4_f32(false, a, false, b, (short)0, c,
                                               false, false);
}

#define ST8(P, V)                                                  \
  do {                                                             \
    float4 lo_, hi_;                                               \
    lo_.x = (V)[0]; lo_.y = (V)[1]; lo_.z = (V)[2]; lo_.w = (V)[3];\
    hi_.x = (V)[4]; hi_.y = (V)[5]; hi_.z = (V)[6]; hi_.w = (V)[7];\
    *(float4*)(P) = lo_;                                           \
    *(float4*)((P) + 4) = hi_;                                     \
  } while (0)

__global__ __launch_bounds__(NT) void lk_flow_kernel(const float* __restrict__ f1,
                                                     const float* __restrict__ f2,
                                                     float* __restrict__ out) {
  extern __shared__ float smem[];
  float* s_Ix  = smem;            // Sobel-x at reflected coords
  float* s_Iy  = smem + PP;       // Sobel-y at reflected coords
  float* s_Itr = smem + 2 * PP;   // It at reflected coords
  float* s_T   = smem + 3 * PP;   // 8 waves * 5 products * TSZ (aliases avg/It)
  float* s_avg = s_T;             // temp: (f1+f2)/2, zero outside image
  float* s_It  = s_T + PP;        // temp: f2-f1 at in-image coords

  const int tid = threadIdx.x;
  const int Y0  = blockIdx.y * 64;   // block output rows Y0..Y0+63
  const int X0  = blockIdx.x * 32;   // block output cols X0..X0+31

  // ---- stage 0: zero ONLY the pad cells the padded WMMA sweeps read ----
  // per gradient plane: rows 79..80 (104 cells) + cols 48..49, rows 1..78 (156)
  for (int i = tid; i < 3 * 260; i += NT) {
    int p = i / 260, j = i % 260;
    float* pl = smem + p * PP;
    if (j < 104) {
      pl[(79 + j / 52) * PW + (j % 52)] = 0.0f;
    } else {
      int jj = j - 104;
      pl[(1 + (jj >> 1)) * PW + 48 + (jj & 1)] = 0.0f;
    }
  }

  // ---- stage 1: frames -> avg/It planes; rows 0..79, cols 1..48 ----
  // column-strip decomposition: 240 threads = 48 cols x 5 row phases, 16 steps
  if (tid < 240) {
    const int lx  = 1 + tid % 48;
    const int r5  = tid / 48;
    const int gx  = X0 - 9 + lx;
    const bool inx = (unsigned)gx < (unsigned)W_IMG;
    int gy = Y0 - 8 + r5;
    int o  = r5 * PW + lx;
    int g  = gy * W_IMG + gx;
#pragma unroll 4
    for (int k = 0; k < 16; ++k) {
      float a = 0.0f, b = 0.0f;
      if (inx && (unsigned)gy < (unsigned)H_IMG) {
        a = f1[g];
        b = f2[g];
      }
      s_avg[o] = 0.5f * (a + b);
      s_It [o] = b - a;
      o  += 5 * PW;
      gy += 5;
      g  += 5 * W_IMG;
    }
  }
  __syncthreads();

  // ---- stage 2: Sobel at reflected coords; rows 1..78, cols 2..47 ----
  // 230 threads = 46 cols x 5 row phases; x-reflection hoisted (col-invariant)
  if (tid < 230) {
    const int lx  = 2 + tid % 46;
    const int r5  = tid / 46;
    const int gx  = X0 - 9 + lx;
    const int lrx = refl(gx) - X0 + 9;
    int ly = 1 + r5;
    int gy = Y0 - 8 + ly;
#pragma unroll 4
    for (int k = 0; k < 16; ++k) {
      if (ly <= 78) {
        int lry = refl(gy) - Y0 + 8;
        const float* c = s_avg + lry * PW + lrx;
        float a00 = c[-PW - 1], a01 = c[-PW], a02 = c[-PW + 1];
        float a10 = c[-1],                    a12 = c[1];
        float a20 = c[PW - 1],  a21 = c[PW],  a22 = c[PW + 1];
        float ix = (a02 - a00) + 2.0f * (a12 - a10) + (a22 - a20);
        float iy = (a20 - a00) + 2.0f * (a21 - a01) + (a22 - a02);
        int o = ly * PW + lx;
        s_Ix[o]  = ix;
        s_Iy[o]  = iy;
        s_Itr[o] = s_It[lry * PW + lrx];
      }
      ly += 5;
      gy += 5;
    }
  }
  __syncthreads();   // after this, avg/It are dead; their space becomes s_T

  // ---- stage 3: per-wave 16x16 tile, separable box sums on the matrix pipe ----
  const int lane  = tid & 31;
  const int wave  = tid >> 5;
  const int tr    = wave >> 1;       // tile row 0..3
  const int tc    = wave & 1;        // tile col 0..1
  const int mrow  = lane & 15;       // per-lane M (and N) index
  const int khalf = lane >> 4;       // 0 -> K {0,1}; 1 -> K {2,3} per chunk
  const int ly0   = 16 * tr + 1;     // plane row of P row 0 for this tile
  const int lx0   = 16 * tc + 2;     // plane col of P col 0 (even!)
  const int cb    = lx0 + 2 * khalf;
  float* Tw = s_T + wave * (5 * TSZ);

  // band constants: shared by pass-1 B (Wh[k][j]) and pass-2 A (Wv[i][k])
  v2f wband[8];
#pragma unroll
  for (int c = 0; c < 8; ++c) {
    int k = 4 * c + 2 * khalf;
    v2f w;
    w.x = bandv(k, mrow);
    w.y = bandv(k + 1, mrow);
    wband[c] = w;
  }

  // pass 1: T_p(32x16) = P_p(32x32, zero-padded) * Wh(32x16), all 5 products fused
#pragma unroll
  for (int m = 0; m < 2; ++m) {
    const int pr = ly0 + 16 * m + mrow;
    const float* rIx = s_Ix  + pr * PW + cb;
    const float* rIy = s_Iy  + pr * PW + cb;
    const float* rIt = s_Itr + pr * PW + cb;
    v8f a0 = {}, a1 = {}, a2 = {}, a3 = {}, a4 = {};
#pragma unroll
    for (int kc = 0; kc < 8; ++kc) {
      float2 ix = *(const float2*)(rIx + 4 * kc);
      float2 iy = *(const float2*)(rIy + 4 * kc);
      float2 it = *(const float2*)(rIt + 4 * kc);
      v2f b = wband[kc], q;
      q.x = ix.x * ix.x; q.y = ix.y * ix.y; a0 = wmma4(q, b, a0);
      q.x = ix.x * iy.x; q.y = ix.y * iy.y; a1 = wmma4(q, b, a1);
      q.x = iy.x * iy.x; q.y = iy.y * iy.y; a2 = wmma4(q, b, a2);
      q.x = ix.x * it.x; q.y = ix.y * it.y; a3 = wmma4(q, b, a3);
      q.x = iy.x * it.x; q.y = iy.y * it.y; a4 = wmma4(q, b, a4);
    }
    // D layout: lane<16 -> M=v, lane>=16 -> M=v+8; stored transposed T[col][row]
    float* tp = Tw + mrow * TS + 16 * m + 8 * khalf;
    ST8(tp + 0 * TSZ, a0);
    ST8(tp + 1 * TSZ, a1);
    ST8(tp + 2 * TSZ, a2);
    ST8(tp + 3 * TSZ, a3);
    ST8(tp + 4 * TSZ, a4);
  }
  __syncthreads();

  // pass 2: Final_p(16x16) = Wv(16x32) * T_p(32x16)
  v8f r0 = {}, r1 = {}, r2 = {}, r3 = {}, r4 = {};
  const float* tb = Tw + mrow * TS + 2 * khalf;   // one base, const offsets below
#pragma unroll
  for (int kc = 0; kc < 8; ++kc) {
    v2f a = wband[kc];
    float2 t;
    v2f b;
    t = *(const float2*)(tb + 0 * TSZ + 4 * kc); b.x = t.x; b.y = t.y; r0 = wmma4(a, b, r0);
    t = *(const float2*)(tb + 1 * TSZ + 4 * kc); b.x = t.x; b.y = t.y; r1 = wmma4(a, b, r1);
    t = *(const float2*)(tb + 2 * TSZ + 4 * kc); b.x = t.x; b.y = t.y; r2 = wmma4(a, b, r2);
    t = *(const float2*)(tb + 3 * TSZ + 4 * kc); b.x = t.x; b.y = t.y; r3 = wmma4(a, b, r3);
    t = *(const float2*)(tb + 4 * TSZ + 4 * kc); b.x = t.x; b.y = t.y; r4 = wmma4(a, b, r4);
  }

  // ---- finalize: per-pixel 2x2 solve, write flow_u then flow_v ----
  const int gxo = X0 + 16 * tc + mrow;
  const int gyb = Y0 + 16 * tr + 8 * khalf;
#pragma unroll
  for (int v = 0; v < 8; ++v) {
    float a00 = r0[v], a01 = r1[v], a11 = r2[v];
    float b0 = -r3[v], b1 = -r4[v];
    float det = a00 * a11 - a01 * a01;
    bool valid = fabsf(det) > 1e-6f;
    float sd = valid ? det : 1.0f;
    float fu = valid ? (a11 * b0 - a01 * b1) / sd : 0.0f;
    float fv = valid ? (a00 * b1 - a01 * b0) / sd : 0.0f;
    int idx = (gyb + v) * W_IMG + gxo;
    out[idx] = fu;
    out[H_IMG * W_IMG + idx] = fv;
  }
}

extern "C" void kernel_launch(void* const* d_in, const int* in_sizes, int n_in,
                              void* d_out, int out_size, void* d_ws, size_t ws_size,
                              hipStream_t stream) {
  (void)in_sizes; (void)n_in; (void)out_size; (void)d_ws; (void)ws_size;
  const float* f1 = (const float*)d_in[0];
  const float* f2 = (const float*)d_in[1];
  float* out = (float*)d_out;
  dim3 grid(W_IMG / 32, H_IMG / 64);   // 64 x 32 blocks, 256 threads each
  size_t shmem = (size_t)(3 * PP + 8 * 5 * TSZ) * sizeof(float);  // 143,328 B
  lk_flow_kernel<<<grid, dim3(NT), shmem, stream>>>(f1, f2, out);
}